// LargeScaleTensorCPFactorization_386547057107
// MI455X (gfx1250) — compile-verified
//
#include <hip/hip_runtime.h>

typedef float v2f __attribute__((ext_vector_type(2)));
typedef float v4f __attribute__((ext_vector_type(4)));
typedef float v8f __attribute__((ext_vector_type(8)));

#define CP_ORDER 4
#define CP_MODES 500000
#define CP_RANK 128

// 4 waves per block, 16 batch rows per wave, rows padded by 4 floats so the
// WMMA-A-layout readback (stride 132 words) hits all 64 LDS banks exactly once.
#define WAVES_PER_BLOCK 4
#define ROWS_PER_WAVE 16
#define LDS_STRIDE (CP_RANK + 4)

__global__ __launch_bounds__(WAVES_PER_BLOCK * 32)
void cp_gather_wmma_kernel(const int* __restrict__ idx,
                           const float* __restrict__ fac,
                           float* __restrict__ out,
                           int batch) {
    __shared__ float tile[WAVES_PER_BLOCK][ROWS_PER_WAVE][LDS_STRIDE];

    const int lane = threadIdx.x & 31;
    // Wave id is uniform across the wave; readfirstlane makes it an SGPR so the
    // index fetches below become scalar loads (broadcast once per wave).
    const int wv   = __builtin_amdgcn_readfirstlane(threadIdx.x >> 5);
    const int base = (blockIdx.x * WAVES_PER_BLOCK + wv) * ROWS_PER_WAVE;

    // ---- Phase 0: hoist all 16 index quads (uniform -> SGPRs, s_load) ----
    int4 ids[ROWS_PER_WAVE];
    #pragma unroll
    for (int r = 0; r < ROWS_PER_WAVE; ++r) {
        int b = base + r;
        int bb = b < batch ? b : batch - 1;   // clamp (never taken for exact grids)
        ids[r] = *(const int4*)(idx + (size_t)bb * CP_ORDER);
    }

    // ---- Phase 1: gather + 4-way elementwise product, coalesced ----
    // Each wave fetches one full 512B factor row per mode per batch row:
    // lane l loads float4 at k = 4*l  (32 lanes * 16B = 512B = whole row).
    const size_t modeStride = (size_t)CP_MODES * CP_RANK;
    const int k = lane << 2;
    #pragma unroll 8
    for (int r = 0; r < ROWS_PER_WAVE; ++r) {
        const float* r0 = fac + (size_t)ids[r].x * CP_RANK;
        const float* r1 = fac + modeStride     + (size_t)ids[r].y * CP_RANK;
        const float* r2 = fac + modeStride * 2 + (size_t)ids[r].z * CP_RANK;
        const float* r3 = fac + modeStride * 3 + (size_t)ids[r].w * CP_RANK;

        v4f v0 = *(const v4f*)(r0 + k);
        v4f v1 = *(const v4f*)(r1 + k);
        v4f v2 = *(const v4f*)(r2 + k);
        v4f v3 = *(const v4f*)(r3 + k);
        v4f g = v0 * v1 * v2 * v3;

        *(v4f*)&tile[wv][r][k] = g;           // ds_store_b128
    }

    __syncthreads();

    // ---- Phase 2: rank reduction via V_WMMA_F32_16X16X4_F32, B = ones ----
    // D[i][j] = sum_k A[i][k] * 1 + C[i][j]  -> every column of D is the row sum.
    // A-layout (16x4 f32): lanes 0-15: M=lane, VGPR0=K0, VGPR1=K1;
    //                      lanes 16-31: M=lane-16, VGPR0=K2, VGPR1=K3.
    const int m = lane & 15;
    const int khalf = (lane >> 4) << 1;       // 0 or 2
    const v2f bones = {1.0f, 1.0f};
    v8f acc = {};

    #pragma unroll
    for (int c = 0; c < CP_RANK / 4; ++c) {
        const int k0 = (c << 2) + khalf;
        v2f a = *(const v2f*)&tile[wv][m][k0];  // ds_load_2addr_b64, conflict-free
        acc = __builtin_amdgcn_wmma_f32_16x16x4_f32(
            /*neg_a=*/false, a, /*neg_b=*/false, bones,
            /*c_mod=*/(short)0, acc, /*reuse_a=*/false, /*reuse_b=*/false);
    }

    // ---- Phase 3: write out column 0 of D ----
    // C/D layout: VGPR v -> M=v (lanes 0-15, N=lane) or M=v+8 (lanes 16-31).
    if (lane == 0 || lane == 16) {
        const int mbase = base + ((lane >> 4) << 3);
        if (mbase + 7 < batch) {
            v4f lo, hi;
            lo[0] = acc[0]; lo[1] = acc[1]; lo[2] = acc[2]; lo[3] = acc[3];
            hi[0] = acc[4]; hi[1] = acc[5]; hi[2] = acc[6]; hi[3] = acc[7];
            *(v4f*)(out + mbase)     = lo;    // global_store_b128
            *(v4f*)(out + mbase + 4) = hi;
        } else {
            #pragma unroll
            for (int i = 0; i < 8; ++i) {
                const int b = mbase + i;
                if (b < batch) out[b] = acc[i];
            }
        }
    }
}

extern "C" void kernel_launch(void* const* d_in, const int* in_sizes, int n_in,
                              void* d_out, int out_size, void* d_ws, size_t ws_size,
                              hipStream_t stream) {
    const int*   idx = (const int*)d_in[0];    // (BATCH, 4) int32
    const float* fac = (const float*)d_in[1];  // (4, 500000, 128) float32
    float*       out = (float*)d_out;          // (BATCH,) float32

    const int batch = out_size;
    const int rows_per_block = WAVES_PER_BLOCK * ROWS_PER_WAVE;  // 64
    const int blocks = (batch + rows_per_block - 1) / rows_per_block;

    cp_gather_wmma_kernel<<<blocks, WAVES_PER_BLOCK * 32, 0, stream>>>(idx, fac, out, batch);
}